// InterClassLoss_16509854286612
// MI455X (gfx1250) — compile-verified
//
#include <hip/hip_runtime.h>
#include <math.h>

#define NUM_CLASSES 19
#define NB 8
#define HWPIX (512 * 1024)
#define EPS_MEAN 1e-6f
#define EPS_PD   1e-6f
#define WS_COLS 40   // cols 0-18: S sums, 19-37: T sums, 38: count, 39: pad
#define WS_FLOATS (NB * NUM_CLASSES * WS_COLS)

typedef float f4  __attribute__((ext_vector_type(4)));
typedef int   i4  __attribute__((ext_vector_type(4)));
typedef float v2f __attribute__((ext_vector_type(2)));
typedef float v8f __attribute__((ext_vector_type(8)));

// ---------------- Kernel 0: zero the workspace accumulator ----------------
__global__ void icl_zero_ws(float* __restrict__ ws) {
  int i = blockIdx.x * blockDim.x + threadIdx.x;
  if (i < WS_FLOATS) ws[i] = 0.0f;
}

// ---------------- Kernel 1: streaming per-class segment sums --------------
#define K1_BLOCKS_PER_IMG 64
#define K1_THREADS 256
#define K1_PIX_PER_BLOCK (HWPIX / K1_BLOCKS_PER_IMG)    // 8192
#define K1_ITERS (K1_PIX_PER_BLOCK / (K1_THREADS * 4))  // 8

__device__ __forceinline__ int clamp_cls(int v) {
  v = v < 0 ? 0 : v;
  return v > (NUM_CLASSES - 1) ? (NUM_CLASSES - 1) : v;
}

__global__ __launch_bounds__(K1_THREADS) void icl_segsum(
    const float* __restrict__ S, const float* __restrict__ T,
    const int* __restrict__ gt, float* __restrict__ ws) {
  __shared__ float acc[NUM_CLASSES][WS_COLS];
  const int tid = threadIdx.x;
  for (int i = tid; i < NUM_CLASSES * WS_COLS; i += K1_THREADS)
    (&acc[0][0])[i] = 0.0f;
  __syncthreads();

  const int n     = blockIdx.x / K1_BLOCKS_PER_IMG;
  const int chunk = blockIdx.x % K1_BLOCKS_PER_IMG;
  const long imgBase = (long)n * HWPIX;

  for (int it = 0; it < K1_ITERS; ++it) {
    const int p = chunk * K1_PIX_PER_BLOCK + it * (K1_THREADS * 4) + tid * 4;
    i4 lab = __builtin_nontemporal_load((const i4*)(gt + imgBase + p));
    const int l0 = clamp_cls(lab.x);
    const int l1 = clamp_cls(lab.y);
    const int l2 = clamp_cls(lab.z);
    const int l3 = clamp_cls(lab.w);
    atomicAdd(&acc[l0][38], 1.0f);
    atomicAdd(&acc[l1][38], 1.0f);
    atomicAdd(&acc[l2][38], 1.0f);
    atomicAdd(&acc[l3][38], 1.0f);
    if (it + 1 < K1_ITERS)
      __builtin_prefetch(gt + imgBase + p + K1_THREADS * 4, 0, 0);

#pragma unroll
    for (int c = 0; c < NUM_CLASSES; ++c) {
      const long off = ((long)n * NUM_CLASSES + c) * HWPIX + p;
      f4 sv = __builtin_nontemporal_load((const f4*)(S + off));
      f4 tv = __builtin_nontemporal_load((const f4*)(T + off));
      if (it + 1 < K1_ITERS) {
        __builtin_prefetch(S + off + K1_THREADS * 4, 0, 0);
        __builtin_prefetch(T + off + K1_THREADS * 4, 0, 0);
      }
      atomicAdd(&acc[l0][c], sv.x);
      atomicAdd(&acc[l1][c], sv.y);
      atomicAdd(&acc[l2][c], sv.z);
      atomicAdd(&acc[l3][c], sv.w);
      atomicAdd(&acc[l0][NUM_CLASSES + c], tv.x);
      atomicAdd(&acc[l1][NUM_CLASSES + c], tv.y);
      atomicAdd(&acc[l2][NUM_CLASSES + c], tv.z);
      atomicAdd(&acc[l3][NUM_CLASSES + c], tv.w);
    }
  }
  __syncthreads();
  for (int i = tid; i < NUM_CLASSES * WS_COLS; i += K1_THREADS) {
    float v = (&acc[0][0])[i];
    if (v != 0.0f)
      atomicAdd(ws + (long)n * NUM_CLASSES * WS_COLS + i, v);
  }
}

// ---------------- Kernel 2: prototypes -> WMMA Gram -> loss ---------------
// e2[i,j] = |V_i|^2 + |V_j - eps|^2 - 2 * (V  (V - eps)^T)[i,j]
// Gram via V_WMMA_F32_16X16X4_F32; classes padded 19->2x16 tiles, K=20.
__global__ __launch_bounds__(256) void icl_loss(const float* __restrict__ ws,
                                                float* __restrict__ out) {
  __shared__ float V[NB][2][32][20];   // [batch][S/T][class row][channel]
  __shared__ float nV[NB][2][32];      // |V_i|^2 over real channels
  __shared__ float nW[NB][2][32];      // |V_j - eps|^2 over real channels
  __shared__ float partial[256];

  const int tid  = threadIdx.x;
  const int w    = tid >> 5;   // wave id == batch index (wave32)
  const int lane = tid & 31;

  // Fill prototypes (lane = class row; rows >= 19 zero-padded)
  if (lane < NUM_CLASSES) {
    const float* row = ws + ((long)w * NUM_CLASSES + lane) * WS_COLS;
    const float inv = 1.0f / (row[38] + EPS_MEAN);
    float ns = 0.f, nt = 0.f, nws = 0.f, nwt = 0.f;
#pragma unroll
    for (int c = 0; c < NUM_CLASSES; ++c) {
      float vs = row[c] * inv;
      float vt = row[NUM_CLASSES + c] * inv;
      V[w][0][lane][c] = vs;
      V[w][1][lane][c] = vt;
      ns += vs * vs;
      nt += vt * vt;
      float wvs = vs - EPS_PD, wvt = vt - EPS_PD;
      nws += wvs * wvs;
      nwt += wvt * wvt;
    }
    V[w][0][lane][19] = 0.0f;  // K padding channel
    V[w][1][lane][19] = 0.0f;
    nV[w][0][lane] = ns;  nV[w][1][lane] = nt;
    nW[w][0][lane] = nws; nW[w][1][lane] = nwt;
  } else {
#pragma unroll
    for (int c = 0; c < 20; ++c) { V[w][0][lane][c] = 0.0f; V[w][1][lane][c] = 0.0f; }
    nV[w][0][lane] = 0.f; nV[w][1][lane] = 0.f;
    nW[w][0][lane] = 0.f; nW[w][1][lane] = 0.f;
  }
  __syncthreads();

  const int mrow = lane & 15;
  const int koff = (lane < 16) ? 0 : 2;  // f32 16x16x4 A/B K mapping
  float part = 0.0f;

#pragma unroll
  for (int ti = 0; ti < 2; ++ti) {
#pragma unroll
    for (int tj = 0; tj < 2; ++tj) {
      v8f accS = {0.f, 0.f, 0.f, 0.f, 0.f, 0.f, 0.f, 0.f};
      v8f accT = {0.f, 0.f, 0.f, 0.f, 0.f, 0.f, 0.f, 0.f};
      const int ar = ti * 16 + mrow;  // A row (class i)
      const int br = tj * 16 + mrow;  // B col (class j)
#pragma unroll
      for (int kc = 0; kc < 5; ++kc) {
        const int ka = 4 * kc + koff;  // channel for this lane's K slots
        v2f aS = *(const v2f*)&V[w][0][ar][ka];
        v2f aT = *(const v2f*)&V[w][1][ar][ka];
        const float e0 = (ka + 0 < NUM_CLASSES) ? EPS_PD : 0.0f;
        const float e1 = (ka + 1 < NUM_CLASSES) ? EPS_PD : 0.0f;
        v2f vb = *(const v2f*)&V[w][0][br][ka];
        v2f bS; bS.x = vb.x - e0; bS.y = vb.y - e1;
        vb = *(const v2f*)&V[w][1][br][ka];
        v2f bT; bT.x = vb.x - e0; bT.y = vb.y - e1;
        accS = __builtin_amdgcn_wmma_f32_16x16x4_f32(
            false, aS, false, bS, (short)0, accS, false, false);
        accT = __builtin_amdgcn_wmma_f32_16x16x4_f32(
            false, aT, false, bT, (short)0, accT, false, false);
      }
      const int j = tj * 16 + mrow;
      const int ibase = ti * 16 + ((lane >= 16) ? 8 : 0);
#pragma unroll
      for (int r = 0; r < 8; ++r) {
        const int i = ibase + r;
        if (i < NUM_CLASSES && j < NUM_CLASSES && j > i) {
          float e2S = nV[w][0][i] + nW[w][0][j] - 2.0f * accS[r];
          float e2T = nV[w][1][i] + nW[w][1][j] - 2.0f * accT[r];
          float eS = sqrtf(fmaxf(e2S, 0.0f));
          float eT = sqrtf(fmaxf(e2T, 0.0f));
          float d = eT - eS;
          part += d * d;
        }
      }
    }
  }

  partial[tid] = part;
  __syncthreads();
  if (tid == 0) {
    float s = 0.0f;
    for (int k = 0; k < 256; ++k) s += partial[k];
    // LOSS_WEIGHT(0.5) * mean over N(8) of sum(0.5 * d^2)  ==  total / 32
    out[0] = s * (1.0f / 32.0f);
  }
}

// ---------------------------- launcher ------------------------------------
extern "C" void kernel_launch(void* const* d_in, const int* in_sizes, int n_in,
                              void* d_out, int out_size, void* d_ws, size_t ws_size,
                              hipStream_t stream) {
  const float* S  = (const float*)d_in[0];
  const float* T  = (const float*)d_in[1];
  const int*   gt = (const int*)d_in[2];
  float* out = (float*)d_out;
  float* ws  = (float*)d_ws;

  icl_zero_ws<<<(WS_FLOATS + 255) / 256, 256, 0, stream>>>(ws);
  icl_segsum<<<NB * K1_BLOCKS_PER_IMG, K1_THREADS, 0, stream>>>(S, T, gt, ws);
  icl_loss<<<1, 256, 0, stream>>>(ws, out);
}